// MCPBRNN_PETconstraint_IBcorrPL_Generic_53695681134748
// MI455X (gfx1250) — compile-verified
//
#include <hip/hip_runtime.h>
#include <math.h>

#define NT      100000
#define GU_N    32
#define U1MAX_F 221.519f
#define ML_F    2.9086f
#define SL_F    1.898f
#define SPIN_I  365
#define TRAIN_I 80000

typedef float v2f __attribute__((ext_vector_type(2)));
typedef float v8f __attribute__((ext_vector_type(8)));

// Static device scratch: g[t]=u1+BC*U1MAX and ol[t], packed for one b64 load
// per serial iteration. Distinct global object => no aliasing with restrict'd
// output pointer, so the compiler can hoist/prefetch loads across the
// loop-carried dependency in the scan kernel.
__device__ float2 g_pre[NT];
__device__ float  g_sd;

__device__ __forceinline__ float fsigmoid(float x) {
    return 1.0f / (1.0f + __expf(-x));
}

// ---------------------------------------------------------------------------
// Kernel 1: BC[t] = relu(u1/U1MAX - bj) @ ln_wj via V_WMMA_F32_16X16X4_F32.
// One wave per 16-timestep tile; 8 chained WMMAs cover K=32 (fp32 exact).
// B has ln_wj replicated into every column -> every column of D equals BC,
// making the result insensitive to N-lane mapping. Also computes ol[t].
// ---------------------------------------------------------------------------
__global__ void __launch_bounds__(32)
k_bc_ol(const float* __restrict__ x,
        const float* __restrict__ w_yom, const float* __restrict__ w_ylm,
        const float* __restrict__ w_yfm,
        const float* __restrict__ b0_ylm, const float* __restrict__ w_b2,
        const float* __restrict__ ln_wj, const float* __restrict__ relu_bj,
        float* __restrict__ out) {
    __shared__ float bcs[16];
    const int lane = threadIdx.x;       // 0..31, wave32
    const int tile = blockIdx.x;        // 6250 tiles * 16 = 100000
    const int m    = lane & 15;         // matrix row = timestep within tile
    const int t    = tile * 16 + m;

    const float u1  = x[2 * t + 0];
    const float u1n = u1 * (1.0f / U1MAX_F);
    // ISA layout for 32-bit A 16x4 / B 4x16: lanes 0-15 hold K={0,1},
    // lanes 16-31 hold K={2,3}; v0 = first, v1 = second of the pair.
    const int klo = (lane < 16) ? 0 : 2;

    v8f acc = {0.f, 0.f, 0.f, 0.f, 0.f, 0.f, 0.f, 0.f};
#pragma unroll
    for (int kb = 0; kb < GU_N; kb += 4) {
        const float bj0 = fmaxf(relu_bj[kb + klo + 0], 0.f);
        const float bj1 = fmaxf(relu_bj[kb + klo + 1], 0.f);
        v2f a, b;
        a.x = fmaxf(u1n - bj0, 0.f);          // A[m][kb+klo]
        a.y = fmaxf(u1n - bj1, 0.f);          // A[m][kb+klo+1]
        b.x = ln_wj[kb + klo + 0];            // B[k][*] = w[k] (all columns)
        b.y = ln_wj[kb + klo + 1];
        acc = __builtin_amdgcn_wmma_f32_16x16x4_f32(
            false, a, false, b, (short)0, acc, false, false);
    }

    // D layout: lanes 0-15 -> M=v, lanes 16-31 -> M=v+8 (all columns equal).
    if (lane == 0) {
#pragma unroll
        for (int v = 0; v < 8; ++v) bcs[v] = acc[v];
    } else if (lane == 16) {
#pragma unroll
        for (int v = 0; v < 8; ++v) bcs[8 + v] = acc[v];
    }
    __syncthreads();

    if (lane < 16) {
        const float BC = bcs[m];
        out[10 * NT + t] = BC;                               // BC_n
        // ol[t] depends only on u2[t] -> fully parallel
        const float eo  = __expf(w_yom[0]);
        const float el  = __expf(w_ylm[0]);
        const float ef  = __expf(w_yfm[0]);
        const float ol1 = el / (eo + el + ef);
        const float u2  = x[2 * t + 1];
        const float ol3 = b0_ylm[0] + (u2 - ML_F) * (1.0f / SL_F) * w_b2[0];
        const float ol  = ol1 * fsigmoid(ol3);
        out[7 * NT + t] = ol;                                // Gate_ol
        float2 pre;
        pre.x = fmaf(BC, U1MAX_F, u1);                       // g[t]
        pre.y = ol;
        g_pre[t] = pre;
    }
}

// ---------------------------------------------------------------------------
// Kernel 2: obs_std = std(y_obs[SPIN:TRAIN], ddof=1). Two-pass, one block.
// ---------------------------------------------------------------------------
__global__ void __launch_bounds__(1024)
k_std(const float* __restrict__ y) {
    __shared__ float red[1024];
    const int tid = threadIdx.x;
    const int n   = TRAIN_I - SPIN_I;   // 79635

    float s = 0.f;
    for (int i = SPIN_I + tid; i < TRAIN_I; i += 1024) s += y[i];
    red[tid] = s;
    __syncthreads();
    for (int off = 512; off > 0; off >>= 1) {
        if (tid < off) red[tid] += red[tid + off];
        __syncthreads();
    }
    const float mean = red[0] / (float)n;
    __syncthreads();

    float ss = 0.f;
    for (int i = SPIN_I + tid; i < TRAIN_I; i += 1024) {
        const float d = y[i] - mean;
        ss += d * d;
    }
    red[tid] = ss;
    __syncthreads();
    for (int off = 512; off > 0; off >>= 1) {
        if (tid < off) red[tid] += red[tid + off];
        __syncthreads();
    }
    if (tid == 0) g_sd = sqrtf(red[0] / (float)(n - 1));
}

// ---------------------------------------------------------------------------
// Kernel 3: zero-fill bp_n / Gate_ib, broadcast obs_std + h_nout column 1.
// ---------------------------------------------------------------------------
__global__ void __launch_bounds__(256)
k_fill(float* __restrict__ out) {
    const int t = blockIdx.x * blockDim.x + threadIdx.x;
    if (t >= NT) return;
    const float sd = g_sd;
    out[ 4 * NT + t]         = 0.f;  // bp_n
    out[ 5 * NT + t]         = 0.f;  // Gate_ib
    out[11 * NT + 2 * t + 1] = sd;   // h_nout[:,1]
    out[13 * NT + t]         = sd;   // obs_std
}

// ---------------------------------------------------------------------------
// Kernel 4: the irreducible serial recurrence. Single lane; minimal chain:
// fma -> v_exp -> rcp -> mul, plus the c0>0 ratio select. All t-only terms
// come prefused from g_pre[t] (one b64 load).
// ---------------------------------------------------------------------------
__global__ void __launch_bounds__(32)
k_scan(const float* __restrict__ x,
       const float* __restrict__ w_yom, const float* __restrict__ w_ylm,
       const float* __restrict__ w_yfm,
       const float* __restrict__ b0_yom, const float* __restrict__ w_b1,
       const float* __restrict__ cmean, const float* __restrict__ cstd,
       float* __restrict__ out) {
    if (threadIdx.x != 0) return;

    const float eo  = __expf(w_yom[0]);
    const float el  = __expf(w_ylm[0]);
    const float ef  = __expf(w_yfm[0]);
    const float oo1 = eo / (eo + el + ef);
    const float Ac  = w_b1[0] / cstd[0];            // oo2 = Ac*c0 + Bc
    const float Bc  = b0_yom[0] - cmean[0] * Ac;

    float c0 = 0.f;
    for (int t = 0; t < NT; ++t) {
        const float2 pre = g_pre[t];                 // {g, ol}
        const float  g   = pre.x;
        const float  ol  = pre.y;
        const float  u2  = x[2 * t + 1];

        const float oo  = oo1 * fsigmoid(fmaf(c0, Ac, Bc));
        float olc = ol;
        if (c0 > 0.f) {
            const float ratio = u2 / c0;
            olc = ol - fmaxf(ol - ratio, 0.f);       // == min(ol, ratio)
        }
        const float f = 1.f - oo - olc;
        const float h = oo * c0;

        out[0 * NT + t]      = h;                    // h_n
        out[1 * NT + t]      = c0;                   // c_n
        out[2 * NT + t]      = ol * c0;              // l_n
        out[3 * NT + t]      = olc * c0;             // lc_n
        out[6 * NT + t]      = oo;                   // Gate_oo
        out[8 * NT + t]      = olc;                  // Gate_olc
        out[9 * NT + t]      = f;                    // Gate_f
        out[11 * NT + 2 * t] = h;                    // h_nout[:,0]

        c0 = fmaf(f, c0, g);
    }
}

// ---------------------------------------------------------------------------
extern "C" void kernel_launch(void* const* d_in, const int* in_sizes, int n_in,
                              void* d_out, int out_size, void* d_ws, size_t ws_size,
                              hipStream_t stream) {
    (void)in_sizes; (void)n_in; (void)out_size; (void)d_ws; (void)ws_size;
    const float* x      = (const float*)d_in[0];
    // d_in[1]=epoch, d_in[2]=time_lag : unused ints
    const float* y_obs  = (const float*)d_in[3];
    const float* cmean  = (const float*)d_in[4];
    const float* cstd   = (const float*)d_in[5];
    const float* w_yom  = (const float*)d_in[6];
    const float* w_ylm  = (const float*)d_in[7];
    const float* w_yfm  = (const float*)d_in[8];
    const float* b0_yom = (const float*)d_in[9];
    const float* w_b1   = (const float*)d_in[10];
    const float* b0_ylm = (const float*)d_in[11];
    const float* w_b2   = (const float*)d_in[12];
    const float* ln_wj  = (const float*)d_in[13];
    const float* relubj = (const float*)d_in[14];
    float* out = (float*)d_out;

    // 1) WMMA matvec (BC) + ol precompute: one wave per 16-timestep tile.
    k_bc_ol<<<NT / 16, 32, 0, stream>>>(x, w_yom, w_ylm, w_yfm,
                                        b0_ylm, w_b2, ln_wj, relubj, out);
    // 2) std(y_obs[365:80000], ddof=1)
    k_std<<<1, 1024, 0, stream>>>(y_obs);
    // 3) fills / broadcasts
    k_fill<<<(NT + 255) / 256, 256, 0, stream>>>(out);
    // 4) irreducible serial scan
    k_scan<<<1, 32, 0, stream>>>(x, w_yom, w_ylm, w_yfm,
                                 b0_yom, w_b1, cmean, cstd, out);
}